// AssumpFreeLoss_23424751633152
// MI455X (gfx1250) — compile-verified
//
#include <hip/hip_runtime.h>
#include <hip/hip_bf16.h>
#include <math.h>

// ---------------------------------------------------------------------------
// AssumpFreeLoss, algebraically collapsed to a single streaming pass over f.
//
//   w_i     = ccp[l_i] / counts[l_i]
//   A       = sum_i w_i * LSE_i                         (scalar)
//   cs[j]   = sum_i w_i * f[i,j]                        (K vector)
//   S[k]    = sum_{i: l_i=k} LSE_i
//   D[k]    = sum_{i: l_i=k} f[i, l_i]
//   diag[k] = counts[k] > 0 ? (S[k]-D[k])/counts[k] : 0
//   lv[j]   = A - cs[j] - (K-1)*ccp[j]*diag[j]
//   out[0]  = sum_j lv[j] ;  out[1+j] = counts[j]*lv[j]
//
// Memory-bound: 524 MB of f once  => ~22.5 us @ 23.3 TB/s.  Data movement
// uses the CDNA5 async global->LDS engine (ASYNCcnt) with per-wave
// double-buffered 4 KB row tiles.
// ---------------------------------------------------------------------------

#define KCLS   1000
#define KPAD   1024
#define NPART  16            // global colsum partial arrays (contention cap)
#define TPB    128           // 4 waves of 32 (wave32)
#define WAVES  4
#define BLOCKS 1024          // 4096 waves -> 32 rows/wave
#define CHUNKS 8             // 8 x 512B per row tile (1024 cols staged)

// ws layout (floats):
//   [0*KPAD .. )  counts
//   [1*KPAD .. )  S   (segment sum of LSE)
//   [2*KPAD .. )  D   (segment sum of f[i, l_i])
//   [3*KPAD]      A
//   [4*KPAD .. )  colsum partials: NPART * KPAD
#define WS_FLOATS ((4 + NPART) * KPAD)

__global__ void afl_init(float* __restrict__ ws, int n) {
    int i = blockIdx.x * blockDim.x + threadIdx.x;
    if (i < n) ws[i] = 0.0f;
}

__global__ void afl_counts(const int* __restrict__ labels,
                           float* __restrict__ counts, int N) {
    int i = blockIdx.x * blockDim.x + threadIdx.x;
    if (i < N) atomicAdd(&counts[labels[i]], 1.0f);
}

__global__ __launch_bounds__(TPB)
void afl_main(const float* __restrict__ f,
              const float* __restrict__ ccp,
              const int*   __restrict__ labels,
              float* __restrict__ ws, int N)
{
    __shared__ __attribute__((aligned(16))) float rowbuf[WAVES][2][KPAD];
    __shared__ float blk_colsum[KPAD];

    float* counts = ws;
    float* Ssum   = ws + 1 * KPAD;
    float* Dsum   = ws + 2 * KPAD;
    float* Aacc   = ws + 3 * KPAD;
    float* part   = ws + 4 * KPAD;

    const int tid  = threadIdx.x;
    const int lane = tid & 31;
    const int wv   = tid >> 5;

    for (int j = tid; j < KPAD; j += TPB) blk_colsum[j] = 0.0f;
    __syncthreads();

    // Per-lane weighted column accumulators: col j = c*128 + lane*4 + q
    float acc[CHUNKS][4];
#pragma unroll
    for (int c = 0; c < CHUNKS; ++c)
#pragma unroll
        for (int q = 0; q < 4; ++q) acc[c][q] = 0.0f;

    const int gw = blockIdx.x * WAVES + wv;
    const int W  = gridDim.x * WAVES;

    // LDS byte offsets of the two per-wave row buffers (flat ptr low 32 bits
    // are the LDS offset on amdgcn shared-aperture addressing).
    const unsigned lds0 = (unsigned)(unsigned long long)(uintptr_t)&rowbuf[wv][0][0];
    const unsigned lds1 = (unsigned)(unsigned long long)(uintptr_t)&rowbuf[wv][1][0];

    float a_local = 0.0f;

    auto consume = [&](int crow, int cp) {
        const float* buf = &rowbuf[wv][cp][0];
        const int   lbl = labels[crow];
        const float w   = ccp[lbl] / counts[lbl];   // counts[lbl] >= 1 here

        float4 v[CHUNKS];
#pragma unroll
        for (int c = 0; c < CHUNKS; ++c)
            v[c] = *(const float4*)&buf[c * 128 + lane * 4];

        // row max (tail chunk valid only for lanes < 26: cols 896..999)
        float m = -3.402823466e+38f;
#pragma unroll
        for (int c = 0; c < CHUNKS - 1; ++c)
            m = fmaxf(m, fmaxf(fmaxf(v[c].x, v[c].y), fmaxf(v[c].z, v[c].w)));
        if (lane < 26) {
            const int c = CHUNKS - 1;
            m = fmaxf(m, fmaxf(fmaxf(v[c].x, v[c].y), fmaxf(v[c].z, v[c].w)));
        }
#pragma unroll
        for (int o = 16; o >= 1; o >>= 1)
            m = fmaxf(m, __shfl_xor(m, o, 32));

        float s = 0.0f;
#pragma unroll
        for (int c = 0; c < CHUNKS - 1; ++c) {
            s += expf(v[c].x - m); s += expf(v[c].y - m);
            s += expf(v[c].z - m); s += expf(v[c].w - m);
        }
        if (lane < 26) {
            const int c = CHUNKS - 1;
            s += expf(v[c].x - m); s += expf(v[c].y - m);
            s += expf(v[c].z - m); s += expf(v[c].w - m);
        }
#pragma unroll
        for (int o = 16; o >= 1; o >>= 1)
            s += __shfl_xor(s, o, 32);

        const float lse = m + logf(s);

        // weighted column accumulation; garbage tail slots (j >= 1000) are
        // finite and never emitted.
#pragma unroll
        for (int c = 0; c < CHUNKS; ++c) {
            acc[c][0] = fmaf(w, v[c].x, acc[c][0]);
            acc[c][1] = fmaf(w, v[c].y, acc[c][1]);
            acc[c][2] = fmaf(w, v[c].z, acc[c][2]);
            acc[c][3] = fmaf(w, v[c].w, acc[c][3]);
        }

        if (lane == 0) {
            a_local += w * lse;
            atomicAdd(&Ssum[lbl], lse);
            atomicAdd(&Dsum[lbl], buf[lbl]);    // f[crow, lbl]
        }
    };

    int prev_row = -1, prev_p = 0, p = 0;
    for (int row = gw; row < N; row += W) {
        // Buffer we are about to overwrite was ds-read in the previous iteration:
        // ensure those LDS reads retired before async writes land.
        asm volatile("s_wait_dscnt 0x0" ::: "memory");

        const unsigned long long gbase =
            (unsigned long long)(uintptr_t)(f + (size_t)row * KCLS);
        const unsigned ldsb = p ? lds1 : lds0;
#pragma unroll
        for (int c = 0; c < CHUNKS; ++c) {
            const int off = c * 512 + lane * 16;                 // bytes
            // chunk 7 lanes >= 26 would read past the 4000-byte row: park
            // them on the (in-bounds) row base; masked in compute.
            const unsigned long long ga =
                (c == CHUNKS - 1 && lane >= 26) ? gbase
                                                : (gbase + (unsigned)off);
            const unsigned la = ldsb + (unsigned)off;
            asm volatile("global_load_async_to_lds_b128 %0, %1, off"
                         :: "v"(la), "v"(ga) : "memory");
        }

        if (prev_row >= 0) {
            // newest 8 async ops (current row) may remain outstanding
            asm volatile("s_wait_asynccnt 0x8" ::: "memory");
            consume(prev_row, prev_p);
        }
        prev_row = row; prev_p = p; p ^= 1;
    }
    if (prev_row >= 0) {
        asm volatile("s_wait_asynccnt 0x0" ::: "memory");
        consume(prev_row, prev_p);
    }

    if (lane == 0 && a_local != 0.0f) atomicAdd(Aacc, a_local);
    if (lane == 0 && a_local == 0.0f && prev_row >= 0) atomicAdd(Aacc, a_local);

    // wave partials -> block LDS colsum -> one of NPART global partials
#pragma unroll
    for (int c = 0; c < CHUNKS; ++c) {
        const int jb = c * 128 + lane * 4;
#pragma unroll
        for (int q = 0; q < 4; ++q) {
            const int j = jb + q;
            if (j < KCLS) atomicAdd(&blk_colsum[j], acc[c][q]);
        }
    }
    __syncthreads();

    float* mypart = part + (size_t)(blockIdx.x % NPART) * KPAD;
    for (int j = tid; j < KCLS; j += TPB)
        atomicAdd(&mypart[j], blk_colsum[j]);
}

__global__ __launch_bounds__(1024)
void afl_finalize(const float* __restrict__ ccp,
                  const float* __restrict__ ws,
                  float* __restrict__ out)
{
    const float* counts = ws;
    const float* Ssum   = ws + 1 * KPAD;
    const float* Dsum   = ws + 2 * KPAD;
    const float  A      = ws[3 * KPAD];
    const float* part   = ws + 4 * KPAD;

    __shared__ float red[32];
    const int tid = threadIdx.x;

    float lv = 0.0f;
    if (tid < KCLS) {
        float cs = 0.0f;
#pragma unroll
        for (int pI = 0; pI < NPART; ++pI) cs += part[pI * KPAD + tid];
        const float cnt  = counts[tid];
        const float diag = (cnt > 0.0f) ? (Ssum[tid] - Dsum[tid]) / cnt : 0.0f;
        lv = A - cs - (float)(KCLS - 1) * ccp[tid] * diag;
        out[1 + tid] = cnt * lv;       // weighted
    }

    float ssum = lv;
#pragma unroll
    for (int o = 16; o >= 1; o >>= 1) ssum += __shfl_xor(ssum, o, 32);
    if ((tid & 31) == 0) red[tid >> 5] = ssum;
    __syncthreads();
    if (tid < 32) {
        float t = red[tid];
#pragma unroll
        for (int o = 16; o >= 1; o >>= 1) t += __shfl_xor(t, o, 32);
        if (tid == 0) out[0] = t;      // final_loss
    }
}

extern "C" void kernel_launch(void* const* d_in, const int* in_sizes, int n_in,
                              void* d_out, int out_size, void* d_ws, size_t ws_size,
                              hipStream_t stream) {
    const float* f      = (const float*)d_in[0];
    const float* ccp    = (const float*)d_in[1];
    const int*   labels = (const int*)d_in[2];
    const int    N      = in_sizes[2];      // 131072; K hardcoded at 1000

    float* ws  = (float*)d_ws;
    float* out = (float*)d_out;

    afl_init  <<<(WS_FLOATS + 255) / 256, 256, 0, stream>>>(ws, WS_FLOATS);
    afl_counts<<<(N + 255) / 256,        256, 0, stream>>>(labels, ws, N);
    afl_main  <<<BLOCKS, TPB,                 0, stream>>>(f, ccp, labels, ws, N);
    afl_finalize<<<1, 1024,                   0, stream>>>(ccp, ws, out);
}